// RotatedRegionProposalNetwork_87686052315910
// MI455X (gfx1250) — compile-verified
//
#include <hip/hip_runtime.h>
#include <math.h>

// ---------------- problem constants ----------------
#define AANCH   3
#define NBATCH  4
#define NLVL    4
#define PRE_N   512
#define POST_N  512
#define NCAND   2048          // 4 levels * 512
#define TANCH   261120
#define NMS_THF 0.7f
#define MIN_SZF 1e-3f
#define LOG_MAXF 4.135166556742356f
#define LVL_OFFF 8192.0f

__device__ constexpr int d_HS[4]   = {256, 128, 64, 32};
__device__ constexpr int d_SIZE[4] = {196608, 49152, 12288, 3072};
__device__ constexpr int d_OFF[4]  = {0, 196608, 245760, 258048};

typedef float v2f __attribute__((ext_vector_type(2)));
typedef float v8f __attribute__((ext_vector_type(8)));

// monotonic float -> uint key (ascending)
__device__ __forceinline__ unsigned fkey(float f) {
    unsigned u = __float_as_uint(f);
    return (u & 0x80000000u) ? ~u : (u | 0x80000000u);
}

// ---------------- kernel 0: zero the suppression bitmask ----------------
__global__ void zero_mask_kernel(unsigned* mask, int nwords) {
    int i = blockIdx.x * blockDim.x + threadIdx.x;
    if (i < nwords) mask[i] = 0u;
}

// ---------------- kernel 1: per-(image,level) top-512 via MSB-first radix select ----
__global__ __launch_bounds__(1024) void topk_kernel(
    const float* __restrict__ obj0, const float* __restrict__ obj1,
    const float* __restrict__ obj2, const float* __restrict__ obj3,
    float* __restrict__ cand_score, int* __restrict__ cand_pos)
{
    const int b = blockIdx.x >> 2, lvl = blockIdx.x & 3;
    const float* obj = (lvl == 0) ? obj0 : (lvl == 1) ? obj1 : (lvl == 2) ? obj2 : obj3;
    const int h = d_HS[lvl], size = d_SIZE[lvl], hw = h * h;
    const float* base = obj + (size_t)b * AANCH * hw;

    __shared__ int hist[256];
    __shared__ unsigned sh_prefix;
    __shared__ int sh_kk, sh_cGT, sh_cEQ;
    const int tid = threadIdx.x, bd = blockDim.x;
    if (tid == 0) { sh_prefix = 0u; sh_kk = PRE_N; }
    __syncthreads();

    for (int pass = 3; pass >= 0; --pass) {
        const int shift = pass * 8;
        for (int t = tid; t < 256; t += bd) hist[t] = 0;
        __syncthreads();
        const unsigned pref = sh_prefix;
        for (int p = tid; p < size; p += bd) {
            const int a = p % AANCH, r = p / AANCH;     // flattened -> [A,H,W] layout
            const unsigned key = fkey(base[a * hw + r]);
            const bool m = (pass == 3) || ((key >> (shift + 8)) == (pref >> (shift + 8)));
            if (m) atomicAdd(&hist[(key >> shift) & 255], 1);
        }
        __syncthreads();
        if (tid == 0) {
            int kk = sh_kk, cum = 0, bsel = 0;
            for (int bin = 255; bin >= 0; --bin) {
                const int c = hist[bin];
                if (cum + c >= kk) { bsel = bin; break; }
                cum += c;
            }
            sh_prefix = pref | ((unsigned)bsel << shift);
            sh_kk = kk - cum;
        }
        __syncthreads();
    }
    const unsigned T = sh_prefix;
    const int needEq = sh_kk, G = PRE_N - needEq;
    if (tid == 0) { sh_cGT = 0; sh_cEQ = 0; }
    __syncthreads();

    const int obase = b * NCAND + lvl * PRE_N;
    for (int p = tid; p < size; p += bd) {
        const int a = p % AANCH, r = p / AANCH;
        const float v = base[a * hw + r];
        const unsigned key = fkey(v);
        int slot = -1;
        if (key > T) slot = atomicAdd(&sh_cGT, 1);
        else if (key == T) {
            const int e = atomicAdd(&sh_cEQ, 1);
            if (e < needEq) slot = G + e;
        }
        if (slot >= 0 && slot < PRE_N) {
            cand_score[obase + slot] = v;
            cand_pos[obase + slot]   = d_OFF[lvl] + p;
        }
    }
}

// ---------------- kernel 2: decode selected boxes + corners/prefilter data --------
__global__ __launch_bounds__(256) void decode_kernel(
    const float* __restrict__ dlt0, const float* __restrict__ dlt1,
    const float* __restrict__ dlt2, const float* __restrict__ dlt3,
    const float* __restrict__ anchors,
    const int* __restrict__ cand_pos,
    float* __restrict__ cand_box, float* __restrict__ cand_corner,
    float* __restrict__ cand_ctr, float* __restrict__ cand_rad,
    float* __restrict__ cand_area, int* __restrict__ cand_valid)
{
    const int idx = blockIdx.x * blockDim.x + threadIdx.x;
    if (idx >= NBATCH * NCAND) return;
    const int b = idx / NCAND, i = idx % NCAND, lvl = i / PRE_N;
    const float* dlt = (lvl == 0) ? dlt0 : (lvl == 1) ? dlt1 : (lvl == 2) ? dlt2 : dlt3;

    const int p = cand_pos[idx];
    const int q = p - d_OFF[lvl];
    const int a = q % AANCH, r = q / AANCH;
    const int h = d_HS[lvl], hw = h * h;
    const size_t dbase = (size_t)b * AANCH * 6 * hw + (size_t)a * 6 * hw + r;
    const float dx = dlt[dbase], dy = dlt[dbase + hw];
    float dw = dlt[dbase + 2 * (size_t)hw], dh = dlt[dbase + 3 * (size_t)hw];
    const float ds = dlt[dbase + 4 * (size_t)hw], dc = dlt[dbase + 5 * (size_t)hw];

    const float* an = anchors + ((size_t)b * TANCH + p) * 5;
    const float ax = an[0], ay = an[1], aw = an[2], ah = an[3], aa = an[4];

    dw = fminf(dw, LOG_MAXF);
    dh = fminf(dh, LOG_MAXF);
    const float cx = ax + dx * aw, cy = ay + dy * ah;
    const float w  = aw * expf(dw), hh = ah * expf(dh);
    const float ang = aa + atan2f(ds, dc);

    cand_box[idx * 5 + 0] = cx; cand_box[idx * 5 + 1] = cy;
    cand_box[idx * 5 + 2] = w;  cand_box[idx * 5 + 3] = hh;
    cand_box[idx * 5 + 4] = ang;
    cand_valid[idx] = (w >= MIN_SZF) && (hh >= MIN_SZF);

    const float off = (float)lvl * LVL_OFFF;
    const float cxo = cx + off, cyo = cy + off;
    cand_ctr[idx * 2]     = cxo;
    cand_ctr[idx * 2 + 1] = cyo;
    cand_area[idx] = w * hh;
    cand_rad[idx]  = 0.5f * sqrtf(w * w + hh * hh);

    float si, co;
    sincosf(ang, &si, &co);
    const float sx[4] = {0.5f, -0.5f, -0.5f, 0.5f};
    const float sy[4] = {0.5f,  0.5f, -0.5f, -0.5f};
#pragma unroll
    for (int k = 0; k < 4; ++k) {
        const float ddx = sx[k] * w, ddy = sy[k] * hh;
        cand_corner[idx * 8 + k]     = cxo + ddx * co - ddy * si;
        cand_corner[idx * 8 + 4 + k] = cyo + ddx * si + ddy * co;
    }
}

// ---------------- kernel 3: bitonic sort 2048 candidates per image (desc) --------
__global__ __launch_bounds__(1024) void sort_kernel(
    const float* __restrict__ cand_score, const float* __restrict__ cand_box,
    const float* __restrict__ cand_corner, const float* __restrict__ cand_ctr,
    const float* __restrict__ cand_rad, const float* __restrict__ cand_area,
    const int* __restrict__ cand_valid,
    float* __restrict__ s_score, float* __restrict__ s_box,
    float* __restrict__ s_corner, float* __restrict__ s_ctr,
    float* __restrict__ s_rad, float* __restrict__ s_area, int* __restrict__ s_valid)
{
    const int b = blockIdx.x, tid = threadIdx.x, bd = blockDim.x;
    __shared__ unsigned sk[NCAND];
    __shared__ int      sv[NCAND];
    for (int i = tid; i < NCAND; i += bd) {
        const int g = b * NCAND + i;
        sk[i] = cand_valid[g] ? fkey(cand_score[g]) : 0u;   // invalid == -inf
        sv[i] = i;
    }
    __syncthreads();
    for (int k = 2; k <= NCAND; k <<= 1) {
        for (int j = k >> 1; j > 0; j >>= 1) {
            for (int i = tid; i < NCAND; i += bd) {
                const int ixj = i ^ j;
                if (ixj > i) {
                    const bool asc = ((i & k) == 0);
                    const unsigned a = sk[i], c = sk[ixj];
                    const bool sw = asc ? (a < c) : (a > c);   // descending overall
                    if (sw) {
                        sk[i] = c; sk[ixj] = a;
                        const int t = sv[i]; sv[i] = sv[ixj]; sv[ixj] = t;
                    }
                }
            }
            __syncthreads();
        }
    }
    for (int i = tid; i < NCAND; i += bd) {
        const int src = b * NCAND + sv[i], dst = b * NCAND + i;
        s_score[dst] = cand_score[src];
#pragma unroll
        for (int c = 0; c < 5; ++c) s_box[dst * 5 + c] = cand_box[src * 5 + c];
#pragma unroll
        for (int c = 0; c < 8; ++c) s_corner[dst * 8 + c] = cand_corner[src * 8 + c];
        s_ctr[dst * 2]     = cand_ctr[src * 2];
        s_ctr[dst * 2 + 1] = cand_ctr[src * 2 + 1];
        s_rad[dst]  = cand_rad[src];
        s_area[dst] = cand_area[src];
        s_valid[dst] = cand_valid[src];
    }
}

// ---------------- rotated-quad intersection (Sutherland-Hodgman) ----------------
__device__ float quad_inter_area(const float* ax, const float* ay,
                                 const float* bx, const float* by)
{
    float px[8], py[8];
#pragma unroll
    for (int i = 0; i < 8; ++i) { px[i] = 0.f; py[i] = 0.f; }
#pragma unroll
    for (int k = 0; k < 4; ++k) { px[k] = ax[k]; py[k] = ay[k]; }
    int cnt = 4;
    for (int k = 0; k < 4; ++k) {
        const float x1 = bx[k], y1 = by[k];
        const float x2 = bx[(k + 1) & 3], y2 = by[(k + 1) & 3];
        const float ex = x2 - x1, ey = y2 - y1;
        float d[8];
        for (int i = 0; i < 8; ++i) d[i] = ex * (py[i] - y1) - ey * (px[i] - x1);
        float qx[8], qy[8];
        int qc = 0;
        for (int i = 0; i < 8; ++i) {
            if (i >= cnt) break;
            const int nx = (i + 1 < cnt) ? i + 1 : 0;
            const bool cin = d[i] >= 0.f, nin = d[nx] >= 0.f;
            if (cin && qc < 8) { qx[qc] = px[i]; qy[qc] = py[i]; ++qc; }
            if ((cin != nin) && qc < 8) {
                const float den = d[i] - d[nx];
                const float t = d[i] / ((den == 0.f) ? 1.f : den);
                qx[qc] = px[i] + t * (px[nx] - px[i]);
                qy[qc] = py[i] + t * (py[nx] - py[i]);
                ++qc;
            }
        }
        cnt = qc;
        for (int i = 0; i < qc; ++i) { px[i] = qx[i]; py[i] = qy[i]; }
        if (cnt == 0) break;
    }
    if (cnt < 3) return 0.f;
    float s = 0.f;
    for (int i = 0; i < 8; ++i) {
        if (i >= cnt) break;
        const int nx = (i + 1 < cnt) ? i + 1 : 0;
        s += px[i] * py[nx] - px[nx] * py[i];
    }
    return 0.5f * fabsf(s);
}

// ---------------- kernel 4: WMMA-prefiltered pairwise IoU -> bitmask -------------
// One wave (32 lanes) per 16x16 tile of (row, col) pairs in score-sorted order.
// V_WMMA_F32_16X16X4_F32 computes the 16x16 center dot-product matrix in one op;
// dist^2 = |ci|^2 + |cj|^2 - 2*dot is the bounding-circle reject test.
__global__ __launch_bounds__(32) void iou_kernel(
    const float* __restrict__ s_corner, const float* __restrict__ s_ctr,
    const float* __restrict__ s_rad, const float* __restrict__ s_area,
    const int* __restrict__ s_valid, unsigned* __restrict__ mask)
{
    const int jt = blockIdx.x, it = blockIdx.y, b = blockIdx.z;
    if (jt < it) return;                 // only need columns j > i
    const int lane = threadIdx.x;

    __shared__ float rX[16][4], rY[16][4], rA[16], rR[16], rCx[16], rCy[16];
    __shared__ float cX[16][4], cY[16][4], cA[16], cR[16], cCx[16], cCy[16];
    __shared__ int   rV[16], cV[16];

    __builtin_prefetch(s_corner + ((size_t)(b * NCAND + jt * 16)) * 8, 0, 0);

    if (lane < 16) {
        const int g = b * NCAND + it * 16 + lane;
#pragma unroll
        for (int k = 0; k < 4; ++k) {
            rX[lane][k] = s_corner[(size_t)g * 8 + k];
            rY[lane][k] = s_corner[(size_t)g * 8 + 4 + k];
        }
        rA[lane] = s_area[g]; rR[lane] = s_rad[g];
        rCx[lane] = s_ctr[2 * (size_t)g]; rCy[lane] = s_ctr[2 * (size_t)g + 1];
        rV[lane] = s_valid[g];
    } else {
        const int n = lane - 16;
        const int g = b * NCAND + jt * 16 + n;
#pragma unroll
        for (int k = 0; k < 4; ++k) {
            cX[n][k] = s_corner[(size_t)g * 8 + k];
            cY[n][k] = s_corner[(size_t)g * 8 + 4 + k];
        }
        cA[n] = s_area[g]; cR[n] = s_rad[g];
        cCx[n] = s_ctr[2 * (size_t)g]; cCy[n] = s_ctr[2 * (size_t)g + 1];
        cV[n] = s_valid[g];
    }
    __syncthreads();

    // A (16x4 f32): lanes 0-15 carry M=lane with K=0,1; lanes 16-31 carry K=2,3 (zero)
    // B (4x16 f32): lanes 0-15 carry N=lane with K=0,1; lanes 16-31 carry K=2,3 (zero)
    v2f av, bv;
    av.x = (lane < 16) ? rCx[lane] : 0.f;
    av.y = (lane < 16) ? rCy[lane] : 0.f;
    bv.x = (lane < 16) ? cCx[lane] : 0.f;
    bv.y = (lane < 16) ? cCy[lane] : 0.f;
    v8f acc = {};
    v8f dots = __builtin_amdgcn_wmma_f32_16x16x4_f32(
        false, av, false, bv, (short)0, acc, false, false);

    const int n  = lane & 15;
    const int mb = (lane >= 16) ? 8 : 0;
    const float cnx = cCx[n], cny = cCy[n];
    const float cn2 = cnx * cnx + cny * cny;
    const float cr = cR[n], ca = cA[n];
    const int cv = cV[n];

#pragma unroll
    for (int v = 0; v < 8; ++v) {
        const int m = v + mb;
        const int gi = it * 16 + m, gj = jt * 16 + n;
        if (gj <= gi) continue;
        if (!rV[m] || !cv) continue;
        const float rmx = rCx[m], rmy = rCy[m];
        const float dist2 = rmx * rmx + rmy * rmy + cn2 - 2.f * dots[v];
        const float rs = rR[m] + cr + 1.0f;               // slack for fp rounding
        if (dist2 >= rs * rs) continue;                    // circles disjoint -> IoU 0
        const float inter = quad_inter_area(rX[m], rY[m], cX[n], cY[n]);
        const float iou = inter / (rA[m] + ca - inter + 1e-7f);
        if (iou > NMS_THF)
            atomicOr(&mask[((size_t)(b * NCAND + gi)) * 64 + (gj >> 5)],
                     1u << (gj & 31));
    }
}

// ---------------- kernel 5: serial suppression sweep + emit top-512 --------------
__global__ __launch_bounds__(64) void nms_out_kernel(
    const float* __restrict__ s_score, const float* __restrict__ s_box,
    const int* __restrict__ s_valid, const unsigned* __restrict__ mask,
    float* __restrict__ out)
{
    const int b = blockIdx.x, tid = threadIdx.x;
    volatile __shared__ unsigned sup[64];
    __shared__ short keep[POST_N];
    __shared__ int nk;

    unsigned w = 0;
    for (int t = 0; t < 32; ++t) {
        const int i = tid * 32 + t;
        if (!s_valid[b * NCAND + i]) w |= 1u << t;         // invalid start suppressed
    }
    sup[tid] = w;
    __syncthreads();

    const unsigned* mrow = mask + (size_t)b * NCAND * 64;
    for (int i = 0; i < NCAND; ++i) {
        const unsigned sw = sup[i >> 5];
        if (!((sw >> (i & 31)) & 1u))                       // i survives -> suppress its row
            sup[tid] |= mrow[(size_t)i * 64 + tid];
        __syncthreads();
    }

    if (tid == 0) {
        int cnt = 0;
        for (int i = 0; i < NCAND && cnt < POST_N; ++i)
            if (!((sup[i >> 5] >> (i & 31)) & 1u)) keep[cnt++] = (short)i;
        nk = cnt;
    }
    __syncthreads();

    float* obox = out;                          // [B, POST_N, 5]
    float* osc  = out + (size_t)NBATCH * POST_N * 5;   // [B, POST_N]
    for (int j = tid; j < POST_N; j += blockDim.x) {
        float bx[5] = {0.f, 0.f, 0.f, 0.f, 0.f};
        float sc = 0.f;
        if (j < nk) {
            const int g = b * NCAND + keep[j];
#pragma unroll
            for (int c = 0; c < 5; ++c) bx[c] = s_box[(size_t)g * 5 + c];
            sc = 1.f / (1.f + expf(-s_score[g]));
        }
#pragma unroll
        for (int c = 0; c < 5; ++c) obox[((size_t)b * POST_N + j) * 5 + c] = bx[c];
        osc[b * POST_N + j] = sc;
    }
}

// ---------------- host launcher ----------------
extern "C" void kernel_launch(void* const* d_in, const int* in_sizes, int n_in,
                              void* d_out, int out_size, void* d_ws, size_t ws_size,
                              hipStream_t stream) {
    // assign inputs by unique element counts (robust to ordering)
    const float *obj[4] = {nullptr, nullptr, nullptr, nullptr};
    const float *dlt[4] = {nullptr, nullptr, nullptr, nullptr};
    const float *anchors = nullptr;
    for (int i = 0; i < n_in; ++i) {
        switch (in_sizes[i]) {
            case 786432:  obj[0] = (const float*)d_in[i]; break;   // 4*3*256*256
            case 196608:  obj[1] = (const float*)d_in[i]; break;   // 4*3*128*128
            case 49152:   obj[2] = (const float*)d_in[i]; break;   // 4*3*64*64
            case 12288:   obj[3] = (const float*)d_in[i]; break;   // 4*3*32*32
            case 4718592: dlt[0] = (const float*)d_in[i]; break;   // 4*18*256*256
            case 1179648: dlt[1] = (const float*)d_in[i]; break;
            case 294912:  dlt[2] = (const float*)d_in[i]; break;
            case 73728:   dlt[3] = (const float*)d_in[i]; break;
            case 5222400: anchors = (const float*)d_in[i]; break;  // 4*261120*5
            default: break;
        }
    }

    // carve workspace
    char* wp = (char*)d_ws;
    auto carve = [&](size_t bytes) {
        void* r = (void*)wp;
        wp += (bytes + 255) & ~(size_t)255;
        return r;
    };
    const size_t NC = (size_t)NBATCH * NCAND;
    float* cand_score  = (float*)carve(NC * 4);
    int*   cand_pos    = (int*)  carve(NC * 4);
    float* cand_box    = (float*)carve(NC * 5 * 4);
    float* cand_corner = (float*)carve(NC * 8 * 4);
    float* cand_ctr    = (float*)carve(NC * 2 * 4);
    float* cand_rad    = (float*)carve(NC * 4);
    float* cand_area   = (float*)carve(NC * 4);
    int*   cand_valid  = (int*)  carve(NC * 4);
    float* s_score     = (float*)carve(NC * 4);
    float* s_box       = (float*)carve(NC * 5 * 4);
    float* s_corner    = (float*)carve(NC * 8 * 4);
    float* s_ctr       = (float*)carve(NC * 2 * 4);
    float* s_rad       = (float*)carve(NC * 4);
    float* s_area      = (float*)carve(NC * 4);
    int*   s_valid     = (int*)  carve(NC * 4);
    unsigned* mask     = (unsigned*)carve(NC * 64 * 4);   // 2 MB bitmask
    (void)ws_size; (void)out_size;

    const int nwords = (int)(NC * 64);
    zero_mask_kernel<<<(nwords + 255) / 256, 256, 0, stream>>>(mask, nwords);

    topk_kernel<<<NBATCH * NLVL, 1024, 0, stream>>>(
        obj[0], obj[1], obj[2], obj[3], cand_score, cand_pos);

    decode_kernel<<<(NBATCH * NCAND + 255) / 256, 256, 0, stream>>>(
        dlt[0], dlt[1], dlt[2], dlt[3], anchors, cand_pos,
        cand_box, cand_corner, cand_ctr, cand_rad, cand_area, cand_valid);

    sort_kernel<<<NBATCH, 1024, 0, stream>>>(
        cand_score, cand_box, cand_corner, cand_ctr, cand_rad, cand_area, cand_valid,
        s_score, s_box, s_corner, s_ctr, s_rad, s_area, s_valid);

    dim3 tg(NCAND / 16, NCAND / 16, NBATCH);   // 128 x 128 x 4 tiles, 1 wave each
    iou_kernel<<<tg, 32, 0, stream>>>(s_corner, s_ctr, s_rad, s_area, s_valid, mask);

    nms_out_kernel<<<NBATCH, 64, 0, stream>>>(s_score, s_box, s_valid, mask,
                                              (float*)d_out);
}